// PointNetSetAbstraction_62646392980106
// MI455X (gfx1250) — compile-verified
//
#include <hip/hip_runtime.h>
#include <math.h>

#define NPOINT   1024
#define NSAMPLE  32
#define RADIUS2  0.04f
#define BATCH    16
#define NPTS     8192
#define EPS_BN   1e-5f
#define RROWS    (BATCH * NPOINT * NSAMPLE)   // 524288 rows into the MLP

typedef __attribute__((ext_vector_type(2))) float v2f;
typedef __attribute__((ext_vector_type(8))) float v8f;

// ---------------------------------------------------------------------------
// 0) init: zero the new_points output region (atomicMax target) + stats accum
// ---------------------------------------------------------------------------
__global__ void init_kernel(float* __restrict__ out_np, float* __restrict__ stats) {
    int i = blockIdx.x * 256 + threadIdx.x;
    if (i < BATCH * NPOINT * 128) out_np[i] = 0.0f;
    if (i < 3 * 256)              stats[i]  = 0.0f;
}

// ---------------------------------------------------------------------------
// 1) Farthest point sampling: one block per batch, serial over 1024 iters.
//    Points live in registers; argmax reduction carries winner coordinates.
// ---------------------------------------------------------------------------
#define FPS_T   512
#define FPS_PPT (NPTS / FPS_T)   // 16 points per thread

__global__ __launch_bounds__(FPS_T)
void fps_kernel(const float* __restrict__ xyz, float* __restrict__ new_xyz) {
    const int b   = blockIdx.x;
    const int tid = threadIdx.x;
    const int lane = tid & 31, wave = tid >> 5;
    const float* base = xyz + (size_t)b * NPTS * 3;

    float px[FPS_PPT], py[FPS_PPT], pz[FPS_PPT], dist[FPS_PPT];
#pragma unroll
    for (int j = 0; j < FPS_PPT; ++j) {
        int p = tid + j * FPS_T;
        px[j] = base[p * 3 + 0];
        py[j] = base[p * 3 + 1];
        pz[j] = base[p * 3 + 2];
        dist[j] = 1e10f;
    }
    // farthest starts at index 0 (reference emits the PRE-update farthest)
    float cx = base[0], cy = base[1], cz = base[2];

    __shared__ float rd[FPS_T / 32], rx[FPS_T / 32], ry[FPS_T / 32], rz[FPS_T / 32];
    __shared__ int   ri[FPS_T / 32];
    __shared__ float bx, by, bz;

    for (int it = 0; it < NPOINT; ++it) {
        if (tid == 0) {
            float* o = new_xyz + ((size_t)b * NPOINT + it) * 3;
            o[0] = cx; o[1] = cy; o[2] = cz;
        }
        // update min-distance, local argmax (with coordinate payload)
        float bd = -1.0f, bxx = 0.f, byy = 0.f, bzz = 0.f;
        int   bp = 0;
#pragma unroll
        for (int j = 0; j < FPS_PPT; ++j) {
            float dx = px[j] - cx, dy = py[j] - cy, dz = pz[j] - cz;
            float d  = dx * dx + dy * dy + dz * dz;
            d = fminf(d, dist[j]);
            dist[j] = d;
            int p = tid + j * FPS_T;
            if (d > bd || (d == bd && p < bp)) {
                bd = d; bp = p; bxx = px[j]; byy = py[j]; bzz = pz[j];
            }
        }
        // wave32 shuffle reduce (argmax, ties -> lowest index like jnp.argmax)
#pragma unroll
        for (int off = 16; off >= 1; off >>= 1) {
            float od = __shfl_xor(bd, off);
            int   op = __shfl_xor(bp, off);
            float ox = __shfl_xor(bxx, off), oy = __shfl_xor(byy, off), oz = __shfl_xor(bzz, off);
            if (od > bd || (od == bd && op < bp)) { bd = od; bp = op; bxx = ox; byy = oy; bzz = oz; }
        }
        if (lane == 0) { rd[wave] = bd; ri[wave] = bp; rx[wave] = bxx; ry[wave] = byy; rz[wave] = bzz; }
        __syncthreads();
        if (wave == 0) {
            const int nw = FPS_T / 32;
            float d2 = (lane < nw) ? rd[lane] : -2.0f;
            int   p2 = (lane < nw) ? ri[lane] : 0;
            float x2 = (lane < nw) ? rx[lane] : 0.f;
            float y2 = (lane < nw) ? ry[lane] : 0.f;
            float z2 = (lane < nw) ? rz[lane] : 0.f;
#pragma unroll
            for (int off = 16; off >= 1; off >>= 1) {
                float od = __shfl_xor(d2, off);
                int   op = __shfl_xor(p2, off);
                float ox = __shfl_xor(x2, off), oy = __shfl_xor(y2, off), oz = __shfl_xor(z2, off);
                if (od > d2 || (od == d2 && op < p2)) { d2 = od; p2 = op; x2 = ox; y2 = oy; z2 = oz; }
            }
            if (lane == 0) { bx = x2; by = y2; bz = z2; }
        }
        __syncthreads();
        cx = bx; cy = by; cz = bz;
    }
}

// ---------------------------------------------------------------------------
// 2) Ball query (lowest-index first 32 within radius) + gather + concat.
//    One wave per (b,s). Writes feat rows of stride 8 (cols 6,7 zero-padded).
// ---------------------------------------------------------------------------
__global__ __launch_bounds__(128)
void ballquery_kernel(const float* __restrict__ xyz, const float* __restrict__ pts,
                      const float* __restrict__ new_xyz, float* __restrict__ feat) {
    const int lane = threadIdx.x & 31;
    const int bs   = blockIdx.x * 4 + (threadIdx.x >> 5);
    const int b    = bs >> 10;

    const float* cp = new_xyz + (size_t)bs * 3;
    const float cx = cp[0], cy = cp[1], cz = cp[2];
    const float* xb = xyz + (size_t)b * NPTS * 3;
    const float* pb = pts + (size_t)b * NPTS * 3;
    float* frow = feat + (size_t)bs * NSAMPLE * 8;

    int  cnt = 0, first_idx = 0;
    bool have_first = false;

    for (int base = 0; base < NPTS; base += 32) {
        int p = base + lane;
        float x = xb[p * 3 + 0], y = xb[p * 3 + 1], z = xb[p * 3 + 2];
        float dx = x - cx, dy = y - cy, dz = z - cz;
        bool in = (dx * dx + dy * dy + dz * dz) <= RADIUS2;
        unsigned mask = (unsigned)__ballot(in);
        if (in) {
            int pos = cnt + __popc(mask & ((1u << lane) - 1u));
            if (pos < NSAMPLE) {
                float* o = frow + (size_t)pos * 8;
                o[0] = dx; o[1] = dy; o[2] = dz;
                o[3] = pb[p * 3 + 0]; o[4] = pb[p * 3 + 1]; o[5] = pb[p * 3 + 2];
                o[6] = 0.f; o[7] = 0.f;
            }
        }
        if (!have_first && mask) { first_idx = base + __ffs((int)mask) - 1; have_first = true; }
        cnt += __popc(mask);
        if (cnt >= NSAMPLE) break;      // cnt is wave-uniform
    }
    int filled = cnt < NSAMPLE ? cnt : NSAMPLE;
    if (filled < NSAMPLE) {             // replicate first in-ball point
        float fx = xb[first_idx * 3 + 0], fy = xb[first_idx * 3 + 1], fz = xb[first_idx * 3 + 2];
        float q0 = pb[first_idx * 3 + 0], q1 = pb[first_idx * 3 + 1], q2 = pb[first_idx * 3 + 2];
        for (int slot = filled + lane; slot < NSAMPLE; slot += 32) {
            float* o = frow + (size_t)slot * 8;
            o[0] = fx - cx; o[1] = fy - cy; o[2] = fz - cz;
            o[3] = q0; o[4] = q1; o[5] = q2; o[6] = 0.f; o[7] = 0.f;
        }
    }
}

// ---------------------------------------------------------------------------
// 3) WMMA f32 GEMM: Y[R x COUT] = act(X)[R x CINP] * W^T + bias
//    - ACT_IN : apply previous layer's BN (batch stats) + ReLU on A-load
//    - STATS  : accumulate per-channel sum/sumsq of raw Y (for this layer's BN)
//    - STORE  : write raw Y
//    - FMAX   : apply this layer's BN + ReLU and atomicMax-reduce over the
//               32 samples of each group into new_points (nonneg-float trick)
//    One wave per 16-row M-tile; V_WMMA_F32_16X16X4_F32, K stepped by 4.
// ---------------------------------------------------------------------------
template<int CINP, int CIN_REAL, int COUT, int NT, int KT,
         bool ACT_IN, bool STORE, bool STATS, bool FMAX>
__global__ __launch_bounds__(256)
void gemm_kernel(const float* __restrict__ X, const float* __restrict__ W,
                 const float* __restrict__ bias,
                 const float* __restrict__ stats_in, const float* __restrict__ g_in,
                 const float* __restrict__ be_in,
                 float* __restrict__ Y, float* __restrict__ stats_out,
                 const float* __restrict__ g_out, const float* __restrict__ be_out,
                 float* __restrict__ out_np) {
    __shared__ __align__(16) float Wl[COUT * CINP];
    __shared__ float biasl[COUT];
    __shared__ float scIn[ACT_IN ? CINP : 1], shIn[ACT_IN ? CINP : 1];
    __shared__ float sAcc[STATS ? COUT : 1],  qAcc[STATS ? COUT : 1];
    __shared__ float scOut[FMAX ? COUT : 1],  shOut[FMAX ? COUT : 1];

    const int tid = threadIdx.x;
    for (int i = tid; i < COUT * CINP; i += 256) {
        int n = i / CINP, k = i % CINP;
        Wl[i] = (k < CIN_REAL) ? W[n * CIN_REAL + k] : 0.0f;
    }
    for (int i = tid; i < COUT; i += 256) {
        biasl[i] = bias[i];
        if (STATS) { sAcc[i] = 0.0f; qAcc[i] = 0.0f; }
        if (FMAX) {
            float mean = stats_out[i] * (1.0f / RROWS);
            float var  = stats_out[128 + i] * (1.0f / RROWS) - mean * mean;
            float sc   = rsqrtf(var + EPS_BN) * g_out[i];
            scOut[i] = sc; shOut[i] = be_out[i] - mean * sc;
        }
    }
    if (ACT_IN) {
        for (int i = tid; i < CINP; i += 256) {
            float mean = stats_in[i] * (1.0f / RROWS);
            float var  = stats_in[128 + i] * (1.0f / RROWS) - mean * mean;
            float sc   = rsqrtf(var + EPS_BN) * g_in[i];
            scIn[i] = sc; shIn[i] = be_in[i] - mean * sc;
        }
    }
    __syncthreads();

    const int lane = tid & 31, lo = lane & 15, hi = lane >> 4;
    const int tile = blockIdx.x * 8 + (tid >> 5);
    const int row0 = tile * 16;

    // A fragments: lane holds X[row0+lo][4t + 2*hi + {0,1}]  (ISA 16x4 f32 layout)
    v2f a[KT];
    const float* xr = X + (size_t)(row0 + lo) * CINP + 2 * hi;
#pragma unroll
    for (int t = 0; t < KT; ++t) {
        v2f v = *(const v2f*)(xr + 4 * t);
        if (ACT_IN) {
            int k = 4 * t + 2 * hi;
            v.x = fmaxf(v.x * scIn[k]     + shIn[k],     0.0f);
            v.y = fmaxf(v.y * scIn[k + 1] + shIn[k + 1], 0.0f);
        }
        a[t] = v;
    }

    for (int nt = 0; nt < NT; ++nt) {
        const int n = nt * 16 + lo;
        float bv = biasl[n];
        v8f c;
#pragma unroll
        for (int i = 0; i < 8; ++i) c[i] = bv;
#pragma unroll
        for (int t = 0; t < KT; ++t) {
            // B fragment: lane holds W[n][4t + 2*hi + {0,1}]  (B = W^T tile)
            v2f bfr = *(const v2f*)&Wl[n * CINP + 4 * t + 2 * hi];
            c = __builtin_amdgcn_wmma_f32_16x16x4_f32(
                    false, a[t], false, bfr, (short)0, c, false, false);
        }
        if (STORE) {
#pragma unroll
            for (int i = 0; i < 8; ++i)
                Y[(size_t)(row0 + i + 8 * hi) * COUT + n] = c[i];
        }
        if (STATS) {
            float s = 0.0f, q = 0.0f;
#pragma unroll
            for (int i = 0; i < 8; ++i) { s += c[i]; q += c[i] * c[i]; }
            atomicAdd(&sAcc[n], s);
            atomicAdd(&qAcc[n], q);
        }
        if (FMAX) {
            float sc = scOut[n], sh = shOut[n];
            float m = 0.0f;   // == ReLU floor; exact since max(relu(v)) = max(0, max v)
#pragma unroll
            for (int i = 0; i < 8; ++i) m = fmaxf(m, c[i] * sc + sh);
            m = fmaxf(m, __shfl_xor(m, 16));       // combine both 8-row halves
            if (hi == 0) {
                int bsI = row0 >> 5;               // tile of 16 rows -> half a 32-group
                atomicMax((int*)(out_np + (size_t)bsI * 128 + n), __float_as_int(m));
            }
        }
    }
    if (STATS) {
        __syncthreads();
        for (int i = tid; i < COUT; i += 256) {
            atomicAdd(&stats_out[i],       sAcc[i]);
            atomicAdd(&stats_out[128 + i], qAcc[i]);
        }
    }
}

// ---------------------------------------------------------------------------
// Host launcher
// ---------------------------------------------------------------------------
extern "C" void kernel_launch(void* const* d_in, const int* in_sizes, int n_in,
                              void* d_out, int out_size, void* d_ws, size_t ws_size,
                              hipStream_t stream) {
    (void)in_sizes; (void)n_in; (void)out_size; (void)ws_size;
    const float* xyz = (const float*)d_in[0];
    const float* pts = (const float*)d_in[1];
    const float* w1 = (const float*)d_in[2],  *b1 = (const float*)d_in[3];
    const float* g1 = (const float*)d_in[4],  *be1 = (const float*)d_in[5];
    const float* w2 = (const float*)d_in[6],  *b2 = (const float*)d_in[7];
    const float* g2 = (const float*)d_in[8],  *be2 = (const float*)d_in[9];
    const float* w3 = (const float*)d_in[10], *b3 = (const float*)d_in[11];
    const float* g3 = (const float*)d_in[12], *be3 = (const float*)d_in[13];

    float* new_xyz = (float*)d_out;                          // (16,1024,3)
    float* out_np  = (float*)d_out + BATCH * NPOINT * 3;     // (16,1024,128)

    float* ws    = (float*)d_ws;
    float* feat  = ws;                                       // R x 8 (padded 6ch)
    float* y1    = feat + (size_t)RROWS * 8;                 // R x 64 raw
    float* y2    = y1   + (size_t)RROWS * 64;                // R x 64 raw
    float* stats = y2   + (size_t)RROWS * 64;                // 3 layers x {sum[128], sumsq[128]}

    init_kernel<<<(BATCH * NPOINT * 128 + 255) / 256, 256, 0, stream>>>(out_np, stats);
    fps_kernel<<<BATCH, FPS_T, 0, stream>>>(xyz, new_xyz);
    ballquery_kernel<<<(BATCH * NPOINT) / 4, 128, 0, stream>>>(xyz, pts, new_xyz, feat);

    const int gblocks = (RROWS / 16) / 8;   // 4096 blocks, 8 waves = 8 M-tiles each

    // layer 1: feat(6,pad8) -> y1(64), collect stats1
    gemm_kernel<8, 6, 64, 4, 2, false, true, true, false>
        <<<gblocks, 256, 0, stream>>>(feat, w1, b1, nullptr, nullptr, nullptr,
                                      y1, stats + 0, nullptr, nullptr, nullptr);
    // layer 2: BN1+ReLU(y1) -> y2(64), collect stats2
    gemm_kernel<64, 64, 64, 4, 16, true, true, true, false>
        <<<gblocks, 256, 0, stream>>>(y1, w2, b2, stats + 0, g1, be1,
                                      y2, stats + 256, nullptr, nullptr, nullptr);
    // layer 3 pass A: BN2+ReLU(y2) -> stats3 only (y3 never materialized)
    gemm_kernel<64, 64, 128, 8, 16, true, false, true, false>
        <<<gblocks, 256, 0, stream>>>(y2, w3, b3, stats + 256, g2, be2,
                                      nullptr, stats + 512, nullptr, nullptr, nullptr);
    // layer 3 pass B: recompute, BN3+ReLU, max over 32 samples -> new_points
    gemm_kernel<64, 64, 128, 8, 16, true, false, false, true>
        <<<gblocks, 256, 0, stream>>>(y2, w3, b3, stats + 256, g2, be2,
                                      nullptr, stats + 512, g3, be3, out_np);
}